// GNNDynamicMemory_3968549782151
// MI455X (gfx1250) — compile-verified
//
#include <hip/hip_runtime.h>
#include <math.h>

typedef float v2f __attribute__((ext_vector_type(2)));
typedef float v8f __attribute__((ext_vector_type(8)));

#define NN 16384   // nodes
#define DD 256     // feature dim
#define MM 256     // hidden dim
#define EE 81920   // candidate edges
#define KK 24576   // kept edges
#define SS 2048    // seq len
#define BB 8       // batch

__device__ __forceinline__ float sigmoidf_(float z) { return 1.0f / (1.0f + expf(-z)); }

// ---------------------------------------------------------------------------
// Edge scoring:  w[e] = sigmoid( relu([x[src],x[dst]] @ E1w + E1b) @ E2w + E2b )
// One wave computes a 16-edge tile with V_WMMA_F32_16X16X4_F32.
// E1w staged in LDS, paired layout: sB[(k>>1)*128 + n*2 + (k&1)]  (512x64 = 128KB)
// ---------------------------------------------------------------------------
__global__ __launch_bounds__(256) void edge_score_kernel(
    const float* __restrict__ x, const int* __restrict__ src, const int* __restrict__ dst,
    const float* __restrict__ E1w, const float* __restrict__ E1b,
    const float* __restrict__ E2w, const float* __restrict__ E2b,
    float* __restrict__ wout)
{
    extern __shared__ float sB[];
    const int tid = threadIdx.x;
    for (int idx = tid; idx < 512 * 64; idx += 256) {
        int k = idx >> 6, n = idx & 63;
        sB[((k >> 1) << 7) + (n << 1) + (k & 1)] = E1w[idx];
    }
    __syncthreads();

    const int wave = tid >> 5, lane = tid & 31;
    const int col = lane & 15, hi = lane >> 4;      // A: row=col, k-offset=2*hi
    const int e0 = (blockIdx.x * 8 + wave) * 16;
    const int e = e0 + col;
    const float* __restrict__ rowS = x + (long)src[e] * DD;
    const float* __restrict__ rowD = x + (long)dst[e] * DD;

    v8f acc0 = {}, acc1 = {}, acc2 = {}, acc3 = {};
#pragma unroll 4
    for (int kk = 0; kk < 128; ++kk) {
        int kbase = kk * 4 + hi * 2;                 // K pair (kbase, kbase+1)
        const float* rp = (kbase < DD) ? (rowS + kbase) : (rowD + (kbase - DD));
        v2f a = *(const v2f*)rp;
        const float* bp = &sB[(kk * 2 + hi) * 128];
        v2f b0 = *(const v2f*)(bp + (col      ) * 2);
        v2f b1 = *(const v2f*)(bp + (col + 16 ) * 2);
        v2f b2 = *(const v2f*)(bp + (col + 32 ) * 2);
        v2f b3 = *(const v2f*)(bp + (col + 48 ) * 2);
        acc0 = __builtin_amdgcn_wmma_f32_16x16x4_f32(false, a, false, b0, (short)0, acc0, false, false);
        acc1 = __builtin_amdgcn_wmma_f32_16x16x4_f32(false, a, false, b1, (short)0, acc1, false, false);
        acc2 = __builtin_amdgcn_wmma_f32_16x16x4_f32(false, a, false, b2, (short)0, acc2, false, false);
        acc3 = __builtin_amdgcn_wmma_f32_16x16x4_f32(false, a, false, b3, (short)0, acc3, false, false);
    }

    // fused bias + relu + (h @ E2w): per lane, columns {col, col+16, col+32, col+48}
    float z[8];
#pragma unroll
    for (int j = 0; j < 8; ++j) z[j] = 0.0f;
#pragma unroll
    for (int nt = 0; nt < 4; ++nt) {
        int c = nt * 16 + col;
        float bb = E1b[c];
        float w2 = E2w[c];
        v8f av = (nt == 0) ? acc0 : ((nt == 1) ? acc1 : ((nt == 2) ? acc2 : acc3));
#pragma unroll
        for (int j = 0; j < 8; ++j) {
            float h = fmaxf(av[j] + bb, 0.0f);
            z[j] += h * w2;
        }
    }
    float b2s = E2b[0];
#pragma unroll
    for (int j = 0; j < 8; ++j) {
        float s = z[j];
        s += __shfl_xor(s, 1, 32);
        s += __shfl_xor(s, 2, 32);
        s += __shfl_xor(s, 4, 32);
        s += __shfl_xor(s, 8, 32);   // each 16-lane half now holds its row sum
        if (lane == 0)  wout[e0 + j]     = sigmoidf_(s + b2s);   // rows 0..7
        if (lane == 16) wout[e0 + 8 + j] = sigmoidf_(s + b2s);   // rows 8..15
    }
}

// ---------------------------------------------------------------------------
// Top-K radix select over positive float bit patterns (sigmoid > 0).
// sel: [0]=hi16 bin, [1]=count above bin, [2]=threshold bits, [3]=ties to take,
//      [4]=tie counter, [5]=kept counter
// ---------------------------------------------------------------------------
__global__ void hist1_kernel(const float* __restrict__ w, unsigned* __restrict__ hist1) {
    int i = blockIdx.x * blockDim.x + threadIdx.x;
    if (i < EE) atomicAdd(&hist1[__float_as_uint(w[i]) >> 16], 1u);
}

__global__ void scan1_kernel(const unsigned* __restrict__ hist1, unsigned* __restrict__ sel) {
    __shared__ unsigned partial[256];
    int t = threadIdx.x;
    unsigned s = 0;
    for (int i = 0; i < 256; ++i) s += hist1[t * 256 + i];
    partial[t] = s;
    __syncthreads();
    if (t == 0) {
        unsigned cum = 0; int coarse = 0;
        for (int c = 255; c >= 0; --c) {
            if (cum + partial[c] >= KK) { coarse = c; break; }
            cum += partial[c];
        }
        unsigned A = cum; int b = coarse * 256;
        for (int f = 255; f >= 0; --f) {
            unsigned h = hist1[coarse * 256 + f];
            if (A + h >= KK) { b = coarse * 256 + f; break; }
            A += h;
        }
        sel[0] = (unsigned)b; sel[1] = A;
    }
}

__global__ void hist2_kernel(const float* __restrict__ w, const unsigned* __restrict__ sel,
                             unsigned* __restrict__ hist2) {
    int i = blockIdx.x * blockDim.x + threadIdx.x;
    if (i < EE) {
        unsigned bits = __float_as_uint(w[i]);
        if ((bits >> 16) == sel[0]) atomicAdd(&hist2[bits & 0xFFFFu], 1u);
    }
}

__global__ void scan2_kernel(const unsigned* __restrict__ hist2, unsigned* __restrict__ sel) {
    __shared__ unsigned partial[256];
    int t = threadIdx.x;
    unsigned s = 0;
    for (int i = 0; i < 256; ++i) s += hist2[t * 256 + i];
    partial[t] = s;
    __syncthreads();
    if (t == 0) {
        unsigned cum = sel[1]; int coarse = 0;
        for (int c = 255; c >= 0; --c) {
            if (cum + partial[c] >= KK) { coarse = c; break; }
            cum += partial[c];
        }
        unsigned A2 = cum; int l = coarse * 256;
        for (int f = 255; f >= 0; --f) {
            unsigned h = hist2[coarse * 256 + f];
            if (A2 + h >= KK) { l = coarse * 256 + f; break; }
            A2 += h;
        }
        sel[2] = (sel[0] << 16) | (unsigned)l;  // exact threshold bits
        sel[3] = KK - A2;                        // ties to take at threshold
    }
}

__global__ void compact_kernel(const float* __restrict__ w, const int* __restrict__ src,
                               const int* __restrict__ dst, unsigned* __restrict__ sel,
                               int* __restrict__ ks, int* __restrict__ kd,
                               unsigned* __restrict__ deg) {
    int i = blockIdx.x * blockDim.x + threadIdx.x;
    if (i >= EE) return;
    unsigned bits = __float_as_uint(w[i]);
    unsigned thr = sel[2];
    bool keep = bits > thr;
    if (!keep && bits == thr) {
        unsigned old = atomicAdd(&sel[4], 1u);
        keep = old < sel[3];
    }
    if (keep) {
        unsigned pos = atomicAdd(&sel[5], 1u);
        ks[pos] = src[i];
        kd[pos] = dst[i];
        atomicAdd(&deg[dst[i]], 1u);
    }
}

__global__ void dinv_kernel(const unsigned* __restrict__ deg, float* __restrict__ dinv) {
    int i = blockIdx.x * blockDim.x + threadIdx.x;
    if (i < NN) dinv[i] = rsqrtf((float)deg[i] + 1.0f);  // +1 for self loop
}

// ---------------------------------------------------------------------------
// hw = A[N,256] @ W[256,64-col panel].  Block = 8 waves; each wave: 16 rows x 64 cols.
// W panel staged in paired LDS layout (256x64 = 64KB).
// ---------------------------------------------------------------------------
__global__ __launch_bounds__(256) void gemm_node_kernel(
    const float* __restrict__ A, const float* __restrict__ W, float* __restrict__ out)
{
    extern __shared__ float sW[];
    const int tid = threadIdx.x;
    const int ny0 = blockIdx.y * 64;
    for (int idx = tid; idx < 256 * 64; idx += 256) {
        int k = idx >> 6, n = idx & 63;
        sW[((k >> 1) << 7) + (n << 1) + (k & 1)] = W[k * MM + ny0 + n];
    }
    __syncthreads();

    const int wave = tid >> 5, lane = tid & 31;
    const int col = lane & 15, hi = lane >> 4;
    const long r0 = (long)blockIdx.x * 128 + wave * 16;
    const float* __restrict__ arow = A + (r0 + col) * DD;

    v8f acc0 = {}, acc1 = {}, acc2 = {}, acc3 = {};
#pragma unroll 4
    for (int kk = 0; kk < 64; ++kk) {
        int kbase = kk * 4 + hi * 2;
        v2f a = *(const v2f*)(arow + kbase);
        const float* bp = &sW[(kk * 2 + hi) * 128];
        v2f b0 = *(const v2f*)(bp + (col      ) * 2);
        v2f b1 = *(const v2f*)(bp + (col + 16 ) * 2);
        v2f b2 = *(const v2f*)(bp + (col + 32 ) * 2);
        v2f b3 = *(const v2f*)(bp + (col + 48 ) * 2);
        acc0 = __builtin_amdgcn_wmma_f32_16x16x4_f32(false, a, false, b0, (short)0, acc0, false, false);
        acc1 = __builtin_amdgcn_wmma_f32_16x16x4_f32(false, a, false, b1, (short)0, acc1, false, false);
        acc2 = __builtin_amdgcn_wmma_f32_16x16x4_f32(false, a, false, b2, (short)0, acc2, false, false);
        acc3 = __builtin_amdgcn_wmma_f32_16x16x4_f32(false, a, false, b3, (short)0, acc3, false, false);
    }
    // C layout: lanes 0-15 -> rows r0+0..7 (VGPR j), lanes 16-31 -> rows r0+8..15
    const long mrow = r0 + hi * 8;
#pragma unroll
    for (int nt = 0; nt < 4; ++nt) {
        v8f av = (nt == 0) ? acc0 : ((nt == 1) ? acc1 : ((nt == 2) ? acc2 : acc3));
        int c = ny0 + nt * 16 + col;
#pragma unroll
        for (int j = 0; j < 8; ++j)
            out[(mrow + j) * MM + c] = av[j];
    }
}

// agg = hw * dinv^2 (self-loop term)
__global__ void self_term_kernel(const float* __restrict__ hw, const float* __restrict__ dinv,
                                 float* __restrict__ agg) {
    long t = (long)blockIdx.x * blockDim.x + threadIdx.x;
    long i = t * 4;
    if (i < (long)NN * MM) {
        float d = dinv[i / MM];
        float d2 = d * d;
        float4 v = *(const float4*)(hw + i);
        float4 r = { v.x * d2, v.y * d2, v.z * d2, v.w * d2 };
        *(float4*)(agg + i) = r;
    }
}

// agg[d] += hw[s] * dinv[s]*dinv[d]  over kept edges (64 threads x float4 per edge)
__global__ void scatter_kernel(const float* __restrict__ hw, const float* __restrict__ dinv,
                               const int* __restrict__ ks, const int* __restrict__ kd,
                               const unsigned* __restrict__ sel, float* __restrict__ agg) {
    long gid = (long)blockIdx.x * blockDim.x + threadIdx.x;
    int e = (int)(gid >> 6);
    int q = (int)(gid & 63);
    if (e >= (int)sel[5]) return;
    int s = ks[e], d = kd[e];
    float norm = dinv[s] * dinv[d];
    int m0 = q * 4;
    float4 v = *(const float4*)(hw + (long)s * MM + m0);
    float* dp = agg + (long)d * MM + m0;
    atomicAdd(dp + 0, v.x * norm);
    atomicAdd(dp + 1, v.y * norm);
    atomicAdd(dp + 2, v.z * norm);
    atomicAdd(dp + 3, v.w * norm);
}

__global__ void bias_act_kernel(const float* __restrict__ agg, const float* __restrict__ b,
                                float* __restrict__ out, int doRelu) {
    long t = (long)blockIdx.x * blockDim.x + threadIdx.x;
    long i = t * 4;
    if (i < (long)NN * MM) {
        int m = (int)(i & (MM - 1));
        float4 v = *(const float4*)(agg + i);
        float4 bb = *(const float4*)(b + m);
        float4 r = { v.x + bb.x, v.y + bb.y, v.z + bb.z, v.w + bb.w };
        if (doRelu) {
            r.x = fmaxf(r.x, 0.0f); r.y = fmaxf(r.y, 0.0f);
            r.z = fmaxf(r.z, 0.0f); r.w = fmaxf(r.w, 0.0f);
        }
        *(float4*)(out + i) = r;
    }
}

// out[b][m] = mean over seq of h2
__global__ void mean_kernel(const float* __restrict__ h2, float* __restrict__ out) {
    int b = blockIdx.x, m = threadIdx.x;
    const float* base = h2 + (long)b * SS * MM + m;
    float s = 0.0f;
    for (int r = 0; r < SS; ++r) s += base[(long)r * MM];
    out[b * MM + m] = s * (1.0f / SS);
}

// feedback = sigmoid(out @ Fw + Fb)
__global__ void feedback_kernel(const float* __restrict__ out, const float* __restrict__ Fw,
                                const float* __restrict__ Fb, float* __restrict__ fb) {
    int d = threadIdx.x;
    for (int b = 0; b < BB; ++b) {
        float z = Fb[d];
        for (int m = 0; m < MM; ++m) z += out[b * MM + m] * Fw[m * DD + d];
        fb[b * DD + d] = sigmoidf_(z);
    }
}

// ---------------------------------------------------------------------------
extern "C" void kernel_launch(void* const* d_in, const int* in_sizes, int n_in,
                              void* d_out, int out_size, void* d_ws, size_t ws_size,
                              hipStream_t stream)
{
    const float* x   = (const float*)d_in[0];
    const int*   src = (const int*)d_in[1];
    const int*   dst = (const int*)d_in[2];
    const float* E1w = (const float*)d_in[3];
    const float* E1b = (const float*)d_in[4];
    const float* E2w = (const float*)d_in[5];
    const float* E2b = (const float*)d_in[6];
    const float* W1  = (const float*)d_in[7];
    const float* b1  = (const float*)d_in[8];
    const float* W2  = (const float*)d_in[9];
    const float* b2  = (const float*)d_in[10];
    const float* Fw  = (const float*)d_in[11];
    const float* Fb  = (const float*)d_in[12];

    char* ws = (char*)d_ws;
    size_t off = 0;
    auto alloc = [&](size_t bytes) {
        void* p = ws + off;
        off += (bytes + 255) & ~(size_t)255;
        return p;
    };
    unsigned* hist1 = (unsigned*)alloc(65536 * 4);
    unsigned* hist2 = (unsigned*)alloc(65536 * 4);
    unsigned* sel   = (unsigned*)alloc(64 * 4);
    unsigned* deg   = (unsigned*)alloc((size_t)NN * 4);
    size_t zeroBytes = off;                    // hist1..deg zeroed every call
    float* wsc  = (float*)alloc((size_t)EE * 4);
    int*   ks   = (int*)alloc((size_t)KK * 4);
    int*   kd   = (int*)alloc((size_t)KK * 4);
    float* dinv = (float*)alloc((size_t)NN * 4);
    float* hw   = (float*)alloc((size_t)NN * MM * 4);
    float* agg  = (float*)alloc((size_t)NN * MM * 4);
    float* hbuf = (float*)alloc((size_t)NN * MM * 4);

    hipMemsetAsync(d_ws, 0, zeroBytes, stream);

    // edge MLP + score
    edge_score_kernel<<<EE / 128, 256, 128 * 1024, stream>>>(x, src, dst, E1w, E1b, E2w, E2b, wsc);

    // top-K selection
    hist1_kernel<<<EE / 256, 256, 0, stream>>>(wsc, hist1);
    scan1_kernel<<<1, 256, 0, stream>>>(hist1, sel);
    hist2_kernel<<<EE / 256, 256, 0, stream>>>(wsc, sel, hist2);
    scan2_kernel<<<1, 256, 0, stream>>>(hist2, sel);
    compact_kernel<<<EE / 256, 256, 0, stream>>>(wsc, src, dst, sel, ks, kd, deg);
    dinv_kernel<<<NN / 256, 256, 0, stream>>>(deg, dinv);

    const long elem4 = (long)NN * MM / 4;
    // GCN layer 1: h1 = relu(norm-agg(x@W1) + b1)
    gemm_node_kernel<<<dim3(NN / 128, MM / 64), 256, 64 * 1024, stream>>>(x, W1, hw);
    self_term_kernel<<<(int)(elem4 / 256), 256, 0, stream>>>(hw, dinv, agg);
    scatter_kernel<<<(KK * 64) / 256, 256, 0, stream>>>(hw, dinv, ks, kd, sel, agg);
    bias_act_kernel<<<(int)(elem4 / 256), 256, 0, stream>>>(agg, b1, hbuf, 1);

    // GCN layer 2: h2 = norm-agg(h1@W2) + b2
    gemm_node_kernel<<<dim3(NN / 128, MM / 64), 256, 64 * 1024, stream>>>(hbuf, W2, hw);
    self_term_kernel<<<(int)(elem4 / 256), 256, 0, stream>>>(hw, dinv, agg);
    scatter_kernel<<<(KK * 64) / 256, 256, 0, stream>>>(hw, dinv, ks, kd, sel, agg);
    bias_act_kernel<<<(int)(elem4 / 256), 256, 0, stream>>>(agg, b2, hw, 0);  // h2 -> hw

    // readout
    float* outp = (float*)d_out;
    mean_kernel<<<BB, MM, 0, stream>>>(hw, outp);
    feedback_kernel<<<1, DD, 0, stream>>>(outp, Fw, Fb, outp + BB * MM);
}